// NMS_7035156431326
// MI455X (gfx1250) — compile-verified
//
#include <hip/hip_runtime.h>
#include <math.h>

typedef __attribute__((ext_vector_type(2))) float v2f;
typedef __attribute__((ext_vector_type(8))) float v8f;

#define GRD 8
#define NSAMP 64
#define LOG2E 1.4426950408889634f

// ---------------- Kernel 1: offset_scale reduction ----------------
__global__ __launch_bounds__(256) void nms_reduce_kernel(const float* __restrict__ boxes,
                                                         int N, float* __restrict__ wsf) {
    __shared__ float sr[256], sc[256];
    int t = threadIdx.x;
    float mr = 0.0f, mc = -1e30f;
    for (int i = t; i < N; i += 256) {
        float w = boxes[i * 5 + 2], h = boxes[i * 5 + 3];
        mr = fmaxf(mr, sqrtf(w * w + h * h));
        mc = fmaxf(mc, fmaxf(boxes[i * 5 + 0], boxes[i * 5 + 1]));
    }
    sr[t] = mr; sc[t] = mc;
    __syncthreads();
    for (int s = 128; s > 0; s >>= 1) {
        if (t < s) { sr[t] = fmaxf(sr[t], sr[t + s]); sc[t] = fmaxf(sc[t], sc[t + s]); }
        __syncthreads();
    }
    if (t == 0) wsf[0] = (sc[0] + 0.5f * sr[0]) * 2.0f + 1.0f;  // offset_scale
}

// ---------------- Kernel 2: stable rank-sort + SoA scatter ----------------
// Stable descending rank = #{j: s_j > s_i} + #{j: s_j == s_i && j < i}
__global__ __launch_bounds__(256) void nms_sort_kernel(const float* __restrict__ boxes,
                                                       const float* __restrict__ scores,
                                                       const int* __restrict__ labels,
                                                       int N, float* __restrict__ wsf) {
    int i = blockIdx.x * 256 + threadIdx.x;
    if (i >= N) return;
    float si = scores[i];
    int rank = 0;
    for (int j = 0; j < N; ++j) {
        float sj = scores[j];
        rank += (sj > si) || (sj == si && j < i);
    }
    float os  = wsf[0];
    float off = (float)labels[i] * os;
    float cx  = boxes[i * 5 + 0] + off;
    float cy  = boxes[i * 5 + 1] + off;
    float w   = boxes[i * 5 + 2];
    float h   = boxes[i * 5 + 3];
    float ang = boxes[i * 5 + 4];
    float c = cosf(ang), s = sinf(ang);

    float* cxs = wsf + 16;
    float* cys = cxs + N;
    float* css = cys + N;
    float* sns = css + N;
    float* hws = sns + N;
    float* hhs = hws + N;
    float* ars = hhs + N;
    float* t1s = ars + N;
    float* t2s = t1s + N;
    float* wss = t2s + N;
    float* hss = wss + N;
    int*   ord = (int*)(hss + N);

    cxs[rank] = cx;          cys[rank] = cy;
    css[rank] = c;           sns[rank] = s;
    hws[rank] = 0.5f * w;    hhs[rank] = 0.5f * h;
    ars[rank] = w * h;
    t1s[rank] = -(c * cx + s * cy);   // tx homogeneous term
    t2s[rank] =  (s * cx - c * cy);   // ty homogeneous term
    wss[rank] = w;           hss[rank] = h;
    ord[rank] = i;
}

// ---------------- Kernel 3: pairwise rotated IoU, one wave per 16x16 tile ----------------
__global__ __launch_bounds__(32) void nms_iou_kernel(const float* __restrict__ wsf,
                                                     float* __restrict__ out, int N) {
    const float* cxs = wsf + 16;
    const float* cys = cxs + N;
    const float* css = cys + N;
    const float* sns = css + N;
    const float* hws = sns + N;
    const float* hhs = hws + N;
    const float* ars = hhs + N;
    const float* t1s = ars + N;
    const float* t2s = t1s + N;
    const float* wss = t2s + N;
    const float* hss = wss + N;

    const int lane = threadIdx.x;
    const int l    = lane & 15;
    const bool hi  = lane >= 16;
    const int i0 = blockIdx.y << 4;
    const int j0 = blockIdx.x << 4;
    const int j  = j0 + l;   // this lane's output column (WMMA C layout: N == lane%16)
    const int i  = i0 + l;   // row owned by lanes 0-15 for A-matrix

    // ---- B/C matrices (4x16, point-invariant). Layout: VGPR0 = rows K=0 (lanes 0-15)
    // and K=2 (lanes 16-31); VGPR1 = rows K=1 / K=3.
    float cj = css[j], sj = sns[j];
    float hwj = hws[j], hhj = hhs[j], aj = ars[j];
    v2f Bm, Cm;
    Bm.x = hi ? t1s[j] : cj;    Bm.y = hi ? 0.0f : sj;
    Cm.x = hi ? t2s[j] : -sj;   Cm.y = hi ? 0.0f : cj;

    // ---- per-row box params for A (lanes 0-15 meaningful).
    // Hi lanes get poisoned params so (px,py) evaluates to the constant
    // homogeneous row (1,0) with NO divergence inside the point loop.
    float cxi = hi ? 1.0f : cxs[i];
    float cyi = hi ? 0.0f : cys[i];
    float ci  = hi ? 0.0f : css[i];
    float si  = hi ? 0.0f : sns[i];
    float wi  = hi ? 0.0f : wss[i];
    float hgt = hi ? 0.0f : hss[i];

    v8f acc = {0, 0, 0, 0, 0, 0, 0, 0};
    const v8f cz = {0, 0, 0, 0, 0, 0, 0, 0};

    for (int p = 0; p < NSAMP; ++p) {
        float ux = ((float)(p >> 3) + 0.5f) * (1.0f / GRD) - 0.5f;
        float uy = ((float)(p & 7)  + 0.5f) * (1.0f / GRD) - 0.5f;
        float lx = ux * wi, ly = uy * hgt;
        float px = cxi + ci * lx - si * ly;   // hi lanes: 1.0
        float py = cyi + si * lx + ci * ly;   // hi lanes: 0.0

        // A (16x4): rows K=0..3 = (px, py, 1, 0). VGPR0: K=0 (lanes 0-15) / K=2 (16-31).
        v2f A;
        A.x = px;
        A.y = py;

        // tx = A*B, ty = A*C  (rank-3 f32 GEMMs on the matrix pipe)
        v8f tx = __builtin_amdgcn_wmma_f32_16x16x4_f32(false, A, false, Bm,
                                                       (short)0, cz, false, false);
        v8f ty = __builtin_amdgcn_wmma_f32_16x16x4_f32(false, A, false, Cm,
                                                       (short)0, cz, false, false);
#pragma unroll
        for (int r = 0; r < 8; ++r) {
            float qx = fabsf(tx[r]) - hwj;
            float qy = fabsf(ty[r]) - hhj;
            float ox = fmaxf(qx, 0.0f), oy = fmaxf(qy, 0.0f);
            float sdf = __builtin_amdgcn_sqrtf(ox * ox + oy * oy + 1e-12f)
                      + fminf(fmaxf(qx, qy), 0.0f);
            // sigmoid(-sdf), SOFT_TAU=1:  1/(1+exp(sdf)) via raw exp2 + rcp
            float e = __builtin_amdgcn_exp2f(sdf * LOG2E);
            acc[r] += __builtin_amdgcn_rcpf(1.0f + e);
        }
    }

#pragma unroll
    for (int r = 0; r < 8; ++r) {
        int gi = i0 + r + (hi ? 8 : 0);              // global row for this C/D register
        float ai    = ars[gi];
        float inter = acc[r] * (1.0f / NSAMP) * ai;
        float uni   = ai + aj - inter;
        float iou   = inter / (uni + 1e-6f);
        if (gi >= j) iou = 0.0f;                     // triu(k=1)
        out[(size_t)gi * N + j] = iou;
    }
}

// ---------------- Kernel 4: keep + order outputs ----------------
__global__ __launch_bounds__(256) void nms_final_kernel(float* __restrict__ out,
                                                        const float* __restrict__ wsf, int N) {
    int j = blockIdx.x * 256 + threadIdx.x;
    if (j >= N) return;
    int cnt = 0;
    for (int i = 0; i < N; ++i) cnt += (out[(size_t)i * N + j] > 0.5f) ? 1 : 0;
    const int* ord = (const int*)(wsf + 16 + 11 * (size_t)N);
    out[(size_t)N * N + j]     = (float)ord[j];              // order
    out[(size_t)N * N + N + j] = (cnt == 0) ? 1.0f : 0.0f;   // keep
}

// ---------------- host launcher ----------------
extern "C" void kernel_launch(void* const* d_in, const int* in_sizes, int n_in,
                              void* d_out, int out_size, void* d_ws, size_t ws_size,
                              hipStream_t stream) {
    const float* boxes  = (const float*)d_in[0];
    const float* scores = (const float*)d_in[1];
    const int*   labels = (const int*)d_in[2];
    int N = in_sizes[1];            // 2048
    float* wsf = (float*)d_ws;
    float* out = (float*)d_out;

    nms_reduce_kernel<<<1, 256, 0, stream>>>(boxes, N, wsf);
    nms_sort_kernel<<<(N + 255) / 256, 256, 0, stream>>>(boxes, scores, labels, N, wsf);
    dim3 grid(N / 16, N / 16);
    nms_iou_kernel<<<grid, 32, 0, stream>>>(wsf, out, N);
    nms_final_kernel<<<(N + 255) / 256, 256, 0, stream>>>(out, wsf, N);
}